// GCNEncoder_86474871537722
// MI455X (gfx1250) — compile-verified
//
#include <hip/hip_runtime.h>

// ---------------------------------------------------------------------------
// 2-layer GCN for MI455X (gfx1250, wave32).
//   layer(x, W, b): g = dinv .* (x @ W)         (WMMA f32 16x16x4, scale fused)
//                   agg seeded = g (self-loop)  (fused into GEMM epilogue)
//                   agg[dst] += g[src]          (f32 atomics, L2-resident)
//                   out = act(dinv .* agg + b)  (finalize)
// FLOP total ~5.4 GF (negligible); runtime is dominated by 2 x 102M f32
// atomics into a 25.6MB buffer that lives in the 192MB L2.
// ---------------------------------------------------------------------------

typedef __attribute__((ext_vector_type(2))) float v2f;
typedef __attribute__((ext_vector_type(8))) float v8f;

#define CH 128  // hidden == out channels (from reference: 128)

// ---------------- degree / dinv --------------------------------------------

__global__ void init_deg_kernel(float* __restrict__ deg, int n) {
  int i = blockIdx.x * blockDim.x + threadIdx.x;
  if (i < n) deg[i] = 1.0f;  // self-loop contributes 1
}

__global__ void edge_deg_kernel(const long long* __restrict__ dst,
                                float* __restrict__ deg, long long nE) {
  long long e = (long long)blockIdx.x * blockDim.x + threadIdx.x;
  if (e < nE) atomicAdd(&deg[(int)dst[e]], 1.0f);
}

__global__ void deg_to_dinv_kernel(float* __restrict__ deg, int n) {
  int i = blockIdx.x * blockDim.x + threadIdx.x;
  if (i < n) deg[i] = rsqrtf(deg[i]);  // deg >= 1 always (self-loops)
}

// ---------------- GEMM: G = AGG = dinv[row] * (A[M,K] @ B[K,128]) ----------
// One block = 256 threads = 8 waves. Block owns 16 rows; wave w owns columns
// [16w, 16w+16). A tile staged in LDS (row pad +2 floats -> bank rotation).
// Fragment layouts per CDNA5 ISA 7.12.2 (32-bit A 16x4 / 32-bit C/D 16x16).
// Epilogue writes both g and the self-loop seed of agg (saves a copy pass).

template <int K>
__global__ void gemm_wmma_scale_kernel(const float* __restrict__ A,
                                       const float* __restrict__ B,
                                       const float* __restrict__ dinv,
                                       float* __restrict__ G,
                                       float* __restrict__ AGG, int M) {
  constexpr int LDK = K + 2;
  __shared__ float lds[16 * LDK];

  const int row0 = blockIdx.x * 16;

  // Cooperative coalesced load of the 16 x K A-tile.
  for (int idx = threadIdx.x; idx < 16 * K; idx += 256) {
    const int r = idx / K;           // constant-folds (K is a power of two)
    const int c = idx & (K - 1);
    const int m = row0 + r;
    lds[r * LDK + c] = (m < M) ? A[(long long)m * K + c] : 0.0f;
  }
  __syncthreads();

  const int wave  = threadIdx.x >> 5;         // 0..7 -> N tile
  const int lane  = threadIdx.x & 31;
  const int c0    = wave * 16;
  const int mLane = lane & 15;
  const int kSel  = (lane < 16) ? 0 : 2;      // A: lanes 16-31 hold K+2,K+3

  v8f acc = {};
#pragma unroll 8
  for (int k0 = 0; k0 < K; k0 += 4) {
    // A fragment: lane(0-15)=row m, VGPR0=K(k0+kSel), VGPR1=K(k0+kSel+1)
    const float* ap = &lds[mLane * LDK + k0 + kSel];
    v2f a;
    a.x = ap[0];
    a.y = ap[1];
    // B fragment (row-major K x 128): lanes 0-15 -> N=c0+lane, rows k0+kSel{,+1}
    const float* bp = &B[(long long)(k0 + kSel) * CH + c0 + mLane];
    v2f b;
    b.x = bp[0];
    b.y = bp[CH];
    acc = __builtin_amdgcn_wmma_f32_16x16x4_f32(
        /*neg_a=*/false, a, /*neg_b=*/false, b,
        /*c_mod=*/(short)0, acc, /*reuse_a=*/false, /*reuse_b=*/false);
  }

  // C/D layout: lane -> N = c0 + (lane&15); VGPR r -> M = r (+8 for hi lanes).
  const int n     = c0 + mLane;
  const int mBase = row0 + ((lane < 16) ? 0 : 8);

  if (row0 + 16 <= M) {
    // Full tile (always true for M % 16 == 0): branch-free epilogue.
    const float4 d0 = *reinterpret_cast<const float4*>(&dinv[mBase]);
    const float4 d1 = *reinterpret_cast<const float4*>(&dinv[mBase + 4]);
    const float dv[8] = {d0.x, d0.y, d0.z, d0.w, d1.x, d1.y, d1.z, d1.w};
    float* gp = &G[(long long)mBase * CH + n];
    float* ap = &AGG[(long long)mBase * CH + n];
#pragma unroll
    for (int r = 0; r < 8; ++r) {
      const float val = dv[r] * acc[r];
      gp[(size_t)(r * CH)] = val;   // immediate offsets r*512B
      ap[(size_t)(r * CH)] = val;
    }
  } else {
#pragma unroll
    for (int r = 0; r < 8; ++r) {
      const int m = mBase + r;
      if (m < M) {
        const float val = dinv[m] * acc[r];
        G[(long long)m * CH + n]   = val;
        AGG[(long long)m * CH + n] = val;
      }
    }
  }
}

// ---------------- edge scatter: agg[dst] += g[src] -------------------------
// One wave (32 lanes) per edge. Edge id is wave-uniform: force it scalar via
// readfirstlane so src[e]/dst[e] become SMEM s_load_b64 and the gather uses a
// scalar base + lane offset (global_load_b128). 4 f32 atomics per lane hit the
// L2-resident agg buffer.

__global__ void scatter_edges_kernel(const long long* __restrict__ src,
                                     const long long* __restrict__ dst,
                                     const float* __restrict__ g,
                                     float* __restrict__ agg, long long nE) {
  const int wid = __builtin_amdgcn_readfirstlane((int)(threadIdx.x >> 5));
  const long long e = (long long)blockIdx.x * 8 + wid;  // blockDim == 256
  if (e >= nE) return;                                  // wave-uniform exit
  const int s = (int)src[e];                            // scalar loads
  const int d = (int)dst[e];
  const int c = ((int)threadIdx.x & 31) * 4;            // 32 lanes x 4 ch = 128
  const float4 v =
      *reinterpret_cast<const float4*>(&g[(long long)s * CH + c]);
  float* o = &agg[(long long)d * CH + c];
  atomicAdd(o + 0, v.x);
  atomicAdd(o + 1, v.y);
  atomicAdd(o + 2, v.z);
  atomicAdd(o + 3, v.w);
}

// ---------------- finalize: out = act(dinv[m]*agg + bias) ------------------

template <bool RELU>
__global__ void finalize_kernel(const float* __restrict__ agg,
                                const float* __restrict__ dinv,
                                const float* __restrict__ bias,
                                float* __restrict__ out, long long n4) {
  long long i = (long long)blockIdx.x * blockDim.x + threadIdx.x;
  if (i >= n4) return;
  const long long base = i * 4;
  const int m = (int)(base >> 7);        // CH == 128
  const int c = (int)(base & (CH - 1));
  const float di = dinv[m];
  float4 v = *reinterpret_cast<const float4*>(&agg[base]);
  const float4 b = *reinterpret_cast<const float4*>(&bias[c]);
  v.x = v.x * di + b.x;
  v.y = v.y * di + b.y;
  v.z = v.z * di + b.z;
  v.w = v.w * di + b.w;
  if (RELU) {
    v.x = fmaxf(v.x, 0.0f);
    v.y = fmaxf(v.y, 0.0f);
    v.z = fmaxf(v.z, 0.0f);
    v.w = fmaxf(v.w, 0.0f);
  }
  *reinterpret_cast<float4*>(&out[base]) = v;
}

// ---------------------------------------------------------------------------

static inline size_t align256(size_t x) { return (x + 255) & ~(size_t)255; }

extern "C" void kernel_launch(void* const* d_in, const int* in_sizes, int n_in,
                              void* d_out, int out_size, void* d_ws,
                              size_t ws_size, hipStream_t stream) {
  (void)n_in; (void)out_size; (void)ws_size;

  const float*     x  = (const float*)d_in[0];
  const long long* ei = (const long long*)d_in[1];
  const float*     W1 = (const float*)d_in[2];
  const float*     b1 = (const float*)d_in[3];
  const float*     W2 = (const float*)d_in[4];
  const float*     b2 = (const float*)d_in[5];

  const int       K1  = 256;
  const int       M   = in_sizes[0] / K1;       // 50000 nodes
  const long long nE  = (long long)in_sizes[1] / 2;
  const long long* src = ei;
  const long long* dst = ei + nE;

  // Workspace carve-up.
  char* ws = (char*)d_ws;
  const size_t featBytes = (size_t)M * CH * sizeof(float);
  float* dinv = (float*)ws;                                      // M floats
  float* g    = (float*)(ws + align256((size_t)M * sizeof(float)));
  float* agg  = (float*)((char*)g + align256(featBytes));
  float* h1   = (float*)((char*)agg + align256(featBytes));
  float* out  = (float*)d_out;

  const long long n4 = (long long)M * CH / 4;
  const int  B256     = 256;
  const int  gNode    = (M + B256 - 1) / B256;
  const int  gEdge    = (int)((nE + B256 - 1) / B256);
  const int  gScatter = (int)((nE + 7) / 8);    // 8 waves (edges) per block
  const int  gVec4    = (int)((n4 + B256 - 1) / B256);
  const int  gGemm    = (M + 15) / 16;

  // --- normalization ---
  init_deg_kernel<<<gNode, B256, 0, stream>>>(dinv, M);
  edge_deg_kernel<<<gEdge, B256, 0, stream>>>(dst, dinv, nE);
  deg_to_dinv_kernel<<<gNode, B256, 0, stream>>>(dinv, M);

  // --- layer 1: g = dinv .* (x @ W1); agg = g + scatter; h1 = relu(...) ---
  gemm_wmma_scale_kernel<256><<<gGemm, B256, 0, stream>>>(x, W1, dinv, g, agg, M);
  scatter_edges_kernel<<<gScatter, B256, 0, stream>>>(src, dst, g, agg, nE);
  finalize_kernel<true><<<gVec4, B256, 0, stream>>>(agg, dinv, b1, h1, n4);

  // --- layer 2: g = dinv .* (h1 @ W2); agg = g + scatter; out = ... ---
  gemm_wmma_scale_kernel<128><<<gGemm, B256, 0, stream>>>(h1, W2, dinv, g, agg, M);
  scatter_edges_kernel<<<gScatter, B256, 0, stream>>>(src, dst, g, agg, nE);
  finalize_kernel<false><<<gVec4, B256, 0, stream>>>(agg, dinv, b2, out, n4);
}